// GCN_attention_v2_11665131176123
// MI455X (gfx1250) — compile-verified
//
#include <hip/hip_runtime.h>
#include <math.h>

#define Nn 6144
#define NFEAT 512
#define NHID 256
#define NCLASS 16

typedef __attribute__((ext_vector_type(2))) float v2f;
typedef __attribute__((ext_vector_type(8))) float v8f;
typedef __attribute__((ext_vector_type(4))) int v4i;

typedef __attribute__((address_space(1))) v4i* gv4ip;  // global (AS1)
typedef __attribute__((address_space(3))) v4i* lv4ip;  // LDS (AS3)

// ---------------------------------------------------------------------------
// CDNA5 async global->LDS copy (GLOBAL_LOAD_ASYNC_TO_LDS_B128, ASYNCcnt).
// Builtin signature (from hipcc diagnostic): (v4i __device__*, v4i __shared__*,
// int offset, int cpol).
// ---------------------------------------------------------------------------
__device__ __forceinline__ void g2l_async_b128(const void* g, void* l) {
#if __has_builtin(__builtin_amdgcn_global_load_async_to_lds_b128)
  __builtin_amdgcn_global_load_async_to_lds_b128((gv4ip)(v4i*)g, (lv4ip)(v4i*)l,
                                                 0, 0);
#else
  const unsigned loff = (unsigned)(uintptr_t)(lv4ip)(v4i*)l;
  const unsigned long long ga = (unsigned long long)(uintptr_t)g;
  asm volatile("global_load_async_to_lds_b128 %0, %1, off"
               :
               : "v"(loff), "v"(ga)
               : "memory");
#endif
}

__device__ __forceinline__ void wait_async0() {
#if __has_builtin(__builtin_amdgcn_s_wait_asynccnt)
  __builtin_amdgcn_s_wait_asynccnt(0);
#else
  asm volatile("s_wait_asynccnt 0x0" ::: "memory");
#endif
}

__device__ __forceinline__ float wredSum(float v) {
  for (int o = 16; o > 0; o >>= 1) v += __shfl_down(v, o, 32);
  return v;
}
__device__ __forceinline__ float wredMax(float v) {
  for (int o = 16; o > 0; o >>= 1) v = fmaxf(v, __shfl_down(v, o, 32));
  return v;
}

// ---------------------------------------------------------------------------
// Gating: z_k = adj_k @ W_at_k + b_at_k (5 cols each), z4 = cat @ W_agg + b,
// nz = softmax3(z4). One block per row. Also writes nj output.
// ---------------------------------------------------------------------------
__global__ void __launch_bounds__(256)
gate_kernel(const float* __restrict__ a0, const float* __restrict__ a1,
            const float* __restrict__ a2,
            const float* __restrict__ W1, const float* __restrict__ b1,
            const float* __restrict__ W2, const float* __restrict__ b2,
            const float* __restrict__ W3, const float* __restrict__ b3,
            const float* __restrict__ Wagg, const float* __restrict__ bagg,
            float* __restrict__ nz, float* __restrict__ nj) {
  const int i = blockIdx.x;
  const int tid = threadIdx.x, lane = tid & 31, w = tid >> 5;
  const float* r0 = a0 + (size_t)i * Nn;
  const float* r1 = a1 + (size_t)i * Nn;
  const float* r2 = a2 + (size_t)i * Nn;
  float s[15];
#pragma unroll
  for (int c = 0; c < 15; ++c) s[c] = 0.f;
  for (int j4 = tid; j4 < Nn / 4; j4 += 256) {
    const int j = j4 << 2;
    const float4 v0 = *(const float4*)(r0 + j);
    const float4 v1 = *(const float4*)(r1 + j);
    const float4 v2 = *(const float4*)(r2 + j);
    const float a0v[4] = {v0.x, v0.y, v0.z, v0.w};
    const float a1v[4] = {v1.x, v1.y, v1.z, v1.w};
    const float a2v[4] = {v2.x, v2.y, v2.z, v2.w};
#pragma unroll
    for (int u = 0; u < 4; ++u) {
      const float* w1 = W1 + (size_t)(j + u) * 5;
      const float* w2 = W2 + (size_t)(j + u) * 5;
      const float* w3 = W3 + (size_t)(j + u) * 5;
#pragma unroll
      for (int c = 0; c < 5; ++c) {
        s[c]      += a0v[u] * w1[c];
        s[5 + c]  += a1v[u] * w2[c];
        s[10 + c] += a2v[u] * w3[c];
      }
    }
  }
  __shared__ float red[15][8];
#pragma unroll
  for (int c = 0; c < 15; ++c) {
    float t = wredSum(s[c]);
    if (lane == 0) red[c][w] = t;
  }
  __syncthreads();
  if (tid == 0) {
    float tot[15];
    for (int c = 0; c < 15; ++c) {
      float t = 0.f;
      for (int q = 0; q < 8; ++q) t += red[c][q];
      tot[c] = t;
    }
    float z4[3];
    for (int m = 0; m < 3; ++m) z4[m] = bagg[m];
    for (int c = 0; c < 5; ++c) {
      const float z1c = tot[c] + b1[c];
      const float z2c = tot[5 + c] + b2[c];
      const float z3c = tot[10 + c] + b3[c];
      for (int m = 0; m < 3; ++m)
        z4[m] += z1c * Wagg[c * 3 + m] + z2c * Wagg[(5 + c) * 3 + m] +
                 z3c * Wagg[(10 + c) * 3 + m];
    }
    const float mx = fmaxf(z4[0], fmaxf(z4[1], z4[2]));
    const float e0 = __expf(z4[0] - mx), e1 = __expf(z4[1] - mx),
                e2 = __expf(z4[2] - mx);
    const float inv = 1.f / (e0 + e1 + e2);
    const float p0 = e0 * inv, p1 = e1 * inv, p2 = e2 * inv;
    nz[i * 3 + 0] = p0; nz[i * 3 + 1] = p1; nz[i * 3 + 2] = p2;
    nj[i * 3 + 0] = p0; nj[i * 3 + 1] = p1; nj[i * 3 + 2] = p2;
  }
}

// ---------------------------------------------------------------------------
// adj[i,j] = sum_k nz[j,k] * adjk[i,j]   (column-wise gating)
// ---------------------------------------------------------------------------
__global__ void __launch_bounds__(256)
fuse_adj_kernel(const float* __restrict__ a0, const float* __restrict__ a1,
                const float* __restrict__ a2, const float* __restrict__ nz,
                float* __restrict__ adj) {
  const size_t total4 = (size_t)Nn * Nn / 4;
  for (size_t q = (size_t)blockIdx.x * blockDim.x + threadIdx.x; q < total4;
       q += (size_t)gridDim.x * blockDim.x) {
    const size_t e = q * 4;
    const int j = (int)(e % Nn);
    const float4 v0 = *(const float4*)(a0 + e);
    const float4 v1 = *(const float4*)(a1 + e);
    const float4 v2 = *(const float4*)(a2 + e);
    float4 o;
    o.x = nz[(j + 0) * 3 + 0] * v0.x + nz[(j + 0) * 3 + 1] * v1.x + nz[(j + 0) * 3 + 2] * v2.x;
    o.y = nz[(j + 1) * 3 + 0] * v0.y + nz[(j + 1) * 3 + 1] * v1.y + nz[(j + 1) * 3 + 2] * v2.y;
    o.z = nz[(j + 2) * 3 + 0] * v0.z + nz[(j + 2) * 3 + 1] * v1.z + nz[(j + 2) * 3 + 2] * v2.z;
    o.w = nz[(j + 3) * 3 + 0] * v0.w + nz[(j + 3) * 3 + 1] * v1.w + nz[(j + 3) * 3 + 2] * v2.w;
    *(float4*)(adj + e) = o;
  }
}

// ---------------------------------------------------------------------------
// FP32 WMMA GEMM, double-buffered LDS fed by GLOBAL_LOAD_ASYNC_TO_LDS_B128.
// C = epilogue(A[M,K] @ op(B)); TRANSB: C = A @ B^T (B is [Nc,K]).
// Block tile 128x64, 8 wave32s, wave tile 32x32 (2x2 V_WMMA_F32_16X16X4_F32).
// LDS strides are 16B multiples (async B128 dest alignment) and bank-spread.
// ---------------------------------------------------------------------------
template <bool TRANSB>
__global__ void __launch_bounds__(256)
gemm_f32_wmma(const float* __restrict__ A, const float* __restrict__ B,
              float* __restrict__ C, int M, int Ncols, int K, int lda, int ldb,
              int ldc, const float* __restrict__ bias,
              const float* __restrict__ rowscale, const float* __restrict__ mask,
              int ldmask, int relu) {
  constexpr int BSR = TRANSB ? 64 : 32;
  constexpr int BSC = TRANSB ? 36 : 68;
  __shared__ float As[2][128][36];   // stride 144B (16B multiple)
  __shared__ float Bs[2][BSR][BSC];  // 36 or 68 floats: 16B-multiple strides
  const int tid = threadIdx.x;
  const int lane = tid & 31, w = tid >> 5;
  const int wm = w >> 1, wn = w & 1;
  const int half = lane >> 4, lm = lane & 15;
  const int m0 = blockIdx.y * 128;
  const int n0 = blockIdx.x * 64;

  v8f acc[2][2] = {};

  auto issue = [&](int k0, int buf) {
#pragma unroll
    for (int i = 0; i < 4; ++i) {  // A tile 128x32: 4 x b128 per thread
      const int id = tid + i * 256;
      const int mr = id >> 3, c4 = (id & 7) << 2;
      g2l_async_b128(A + (size_t)(m0 + mr) * lda + k0 + c4, &As[buf][mr][c4]);
    }
    if (!TRANSB) {  // B tile 32x64, k-major rows
#pragma unroll
      for (int i = 0; i < 2; ++i) {
        const int id = tid + i * 256;
        const int kr = id >> 4, c4 = (id & 15) << 2;
        if (n0 + c4 < Ncols)
          g2l_async_b128(B + (size_t)(k0 + kr) * ldb + n0 + c4,
                         &Bs[buf][kr][c4]);
      }
    } else {  // B tile 64 rows (n) x 32 (k), k-contiguous like A
#pragma unroll
      for (int i = 0; i < 2; ++i) {
        const int id = tid + i * 256;
        const int nr = id >> 3, k4 = (id & 7) << 2;
        if (n0 + nr < Ncols)
          g2l_async_b128(B + (size_t)(n0 + nr) * ldb + k0 + k4,
                         &Bs[buf][nr][k4]);
      }
    }
  };

  const int nst = K / 32;
  issue(0, 0);
  for (int s = 0; s < nst; ++s) {
    wait_async0();     // own tile-s async writes to LDS complete
    __syncthreads();   // all waves' tile-s data visible; all done computing s-1
    if (s + 1 < nst) issue((s + 1) * 32, (s + 1) & 1);  // overlap with compute
    const int cb = s & 1;
#pragma unroll
    for (int kk = 0; kk < 32; kk += 4) {
      const int kc = kk + half * 2;  // A[m,k]: lane = m + 16*(k>=2), vgpr = k&1
      v2f a0q, a1q, b0q, b1q;
      a0q.x = As[cb][wm * 32 + lm][kc];      a0q.y = As[cb][wm * 32 + lm][kc + 1];
      a1q.x = As[cb][wm * 32 + 16 + lm][kc]; a1q.y = As[cb][wm * 32 + 16 + lm][kc + 1];
      if (TRANSB) {
        b0q.x = Bs[cb][wn * 32 + lm][kc];      b0q.y = Bs[cb][wn * 32 + lm][kc + 1];
        b1q.x = Bs[cb][wn * 32 + 16 + lm][kc]; b1q.y = Bs[cb][wn * 32 + 16 + lm][kc + 1];
      } else {
        b0q.x = Bs[cb][kc][wn * 32 + lm];      b0q.y = Bs[cb][kc + 1][wn * 32 + lm];
        b1q.x = Bs[cb][kc][wn * 32 + 16 + lm]; b1q.y = Bs[cb][kc + 1][wn * 32 + 16 + lm];
      }
      acc[0][0] = __builtin_amdgcn_wmma_f32_16x16x4_f32(
          false, a0q, false, b0q, (short)0, acc[0][0], false, false);
      acc[0][1] = __builtin_amdgcn_wmma_f32_16x16x4_f32(
          false, a0q, false, b1q, (short)0, acc[0][1], false, false);
      acc[1][0] = __builtin_amdgcn_wmma_f32_16x16x4_f32(
          false, a1q, false, b0q, (short)0, acc[1][0], false, false);
      acc[1][1] = __builtin_amdgcn_wmma_f32_16x16x4_f32(
          false, a1q, false, b1q, (short)0, acc[1][1], false, false);
    }
  }
  // epilogue: C/D layout vgpr r <-> M = r + 8*half, N = lane&15
#pragma unroll
  for (int mi = 0; mi < 2; ++mi)
#pragma unroll
    for (int ni = 0; ni < 2; ++ni) {
      const int col = n0 + wn * 32 + ni * 16 + lm;
      if (col >= Ncols) continue;
      const float bcol = bias ? bias[col] : 0.f;
#pragma unroll
      for (int r = 0; r < 8; ++r) {
        const int row = m0 + wm * 32 + mi * 16 + r + 8 * half;
        float v = acc[mi][ni][r] + bcol;
        if (mask) v *= mask[(size_t)row * ldmask + col];
        if (rowscale) v *= rowscale[row];
        if (relu) v = fmaxf(v, 0.f);
        C[(size_t)row * ldc + col] = v;
      }
    }
}

// ---------------------------------------------------------------------------
// Row softmax (in place) + dis = rsqrt(rowsum(attention)); gcn_norm folds to
// dis[i]*att*dis[j] since the softmax rowsum feeds both factors.
// ---------------------------------------------------------------------------
__global__ void __launch_bounds__(256)
row_softmax_kernel(float* __restrict__ At, float* __restrict__ dis) {
  const int i = blockIdx.x;
  float* row = At + (size_t)i * Nn;
  __shared__ float sred[8];
  __shared__ float sbc;
  const int tid = threadIdx.x, lane = tid & 31, w = tid >> 5;

  float m = -__builtin_inff();
  for (int j = tid; j < Nn; j += 256) m = fmaxf(m, row[j]);
  m = wredMax(m);
  if (lane == 0) sred[w] = m;
  __syncthreads();
  if (tid == 0) {
    float mm = sred[0];
    for (int q = 1; q < 8; ++q) mm = fmaxf(mm, sred[q]);
    sbc = mm;
  }
  __syncthreads();
  m = sbc;
  __syncthreads();

  float s = 0.f;
  for (int j = tid; j < Nn; j += 256) s += __expf(row[j] - m);
  s = wredSum(s);
  if (lane == 0) sred[w] = s;
  __syncthreads();
  if (tid == 0) {
    float ss = 0.f;
    for (int q = 0; q < 8; ++q) ss += sred[q];
    sbc = ss;
  }
  __syncthreads();
  const float inv = 1.f / sbc;
  __syncthreads();

  float t = 0.f;
  for (int j = tid; j < Nn; j += 256) {
    const float r = __expf(row[j] - m) * inv;
    row[j] = r;
    t += r;
  }
  t = wredSum(t);
  if (lane == 0) sred[w] = t;
  __syncthreads();
  if (tid == 0) {
    float d = 0.f;
    for (int q = 0; q < 8; ++q) d += sred[q];
    dis[i] = (d > 0.f) ? rsqrtf(d) : 0.f;
  }
}

__global__ void __launch_bounds__(256)
scale_rows_kernel(float* __restrict__ V, const float* __restrict__ dis) {
  const int idx = blockIdx.x * blockDim.x + threadIdx.x;
  V[idx] *= dis[idx >> 8];  // NHID == 256
}

__global__ void __launch_bounds__(256)
softmax16_kernel(const float* __restrict__ z, float* __restrict__ out) {
  const int i = blockIdx.x * blockDim.x + threadIdx.x;
  if (i >= Nn) return;
  const float* r = z + (size_t)i * NCLASS;
  float mx = r[0];
#pragma unroll
  for (int c = 1; c < NCLASS; ++c) mx = fmaxf(mx, r[c]);
  float e[NCLASS], s = 0.f;
#pragma unroll
  for (int c = 0; c < NCLASS; ++c) { e[c] = __expf(r[c] - mx); s += e[c]; }
  const float inv = 1.f / s;
#pragma unroll
  for (int c = 0; c < NCLASS; ++c) out[(size_t)i * NCLASS + c] = e[c] * inv;
}

// ---------------------------------------------------------------------------
extern "C" void kernel_launch(void* const* d_in, const int* in_sizes, int n_in,
                              void* d_out, int out_size, void* d_ws,
                              size_t ws_size, hipStream_t stream) {
  const float* adj0 = (const float*)d_in[0];
  const float* adj1 = (const float*)d_in[1];
  const float* adj2 = (const float*)d_in[2];
  const float* x    = (const float*)d_in[3];
  const float* W_at1 = (const float*)d_in[4];
  const float* b_at1 = (const float*)d_in[5];
  const float* W_at2 = (const float*)d_in[6];
  const float* b_at2 = (const float*)d_in[7];
  const float* W_at3 = (const float*)d_in[8];
  const float* b_at3 = (const float*)d_in[9];
  const float* W_agg = (const float*)d_in[10];
  const float* b_agg = (const float*)d_in[11];
  const float* W_g1 = (const float*)d_in[12];
  const float* b_g1 = (const float*)d_in[13];
  const float* W_g2 = (const float*)d_in[14];
  const float* b_g2 = (const float*)d_in[15];
  const float* W_q = (const float*)d_in[16];
  const float* b_q = (const float*)d_in[17];
  const float* W_k = (const float*)d_in[18];
  const float* b_k = (const float*)d_in[19];
  const float* W_v = (const float*)d_in[20];
  const float* b_v = (const float*)d_in[21];

  float* ws = (float*)d_ws;
  const size_t NNsz = (size_t)Nn * Nn;
  float* adj = ws;
  float* At  = adj + NNsz;
  float* xw  = At + NNsz;
  float* h   = xw + (size_t)Nn * NHID;
  float* Q   = h + (size_t)Nn * NHID;
  float* Kh  = Q + (size_t)Nn * NHID;
  float* V   = Kh + (size_t)Nn * NHID;
  float* Xt  = V + (size_t)Nn * NHID;
  float* xw2 = Xt + (size_t)Nn * NHID;
  float* nzb = xw2 + (size_t)Nn * NCLASS;
  float* dis = nzb + (size_t)Nn * 3;
  float* zbuf = xw;  // xw dead after h-GEMM; reuse for z [N,16]

  float* out_probs = (float*)d_out;
  float* out_nj = out_probs + (size_t)Nn * NCLASS;

  const dim3 blk(256);

  gate_kernel<<<Nn, blk, 0, stream>>>(adj0, adj1, adj2, W_at1, b_at1, W_at2,
                                      b_at2, W_at3, b_at3, W_agg, b_agg, nzb,
                                      out_nj);
  fuse_adj_kernel<<<6144, blk, 0, stream>>>(adj0, adj1, adj2, nzb, adj);
  // xw = x @ W_g1
  gemm_f32_wmma<false><<<dim3(NHID / 64, Nn / 128), blk, 0, stream>>>(
      x, W_g1, xw, Nn, NHID, NFEAT, NFEAT, NHID, NHID, nullptr, nullptr,
      nullptr, 0, 0);
  // h = relu(adj @ xw + b_g1)
  gemm_f32_wmma<false><<<dim3(NHID / 64, Nn / 128), blk, 0, stream>>>(
      adj, xw, h, Nn, NHID, Nn, Nn, NHID, NHID, b_g1, nullptr, nullptr, 0, 1);
  // Q, K, V
  gemm_f32_wmma<false><<<dim3(NHID / 64, Nn / 128), blk, 0, stream>>>(
      h, W_q, Q, Nn, NHID, NHID, NHID, NHID, NHID, b_q, nullptr, nullptr, 0, 0);
  gemm_f32_wmma<false><<<dim3(NHID / 64, Nn / 128), blk, 0, stream>>>(
      h, W_k, Kh, Nn, NHID, NHID, NHID, NHID, NHID, b_k, nullptr, nullptr, 0, 0);
  gemm_f32_wmma<false><<<dim3(NHID / 64, Nn / 128), blk, 0, stream>>>(
      h, W_v, V, Nn, NHID, NHID, NHID, NHID, NHID, b_v, nullptr, nullptr, 0, 0);
  // A_tilde = adj * (Q @ Kh^T)
  gemm_f32_wmma<true><<<dim3(Nn / 64, Nn / 128), blk, 0, stream>>>(
      Q, Kh, At, Nn, Nn, NHID, NHID, NHID, Nn, nullptr, nullptr, adj, Nn, 0);
  // attention = row-softmax(A_tilde); dis = rsqrt(rowsum)
  row_softmax_kernel<<<Nn, blk, 0, stream>>>(At, dis);
  // V[j,:] *= dis[j]
  scale_rows_kernel<<<(Nn * NHID) / 256, blk, 0, stream>>>(V, dis);
  // X_tilde = relu(dis[i] * (attention @ V'))
  gemm_f32_wmma<false><<<dim3(NHID / 64, Nn / 128), blk, 0, stream>>>(
      At, V, Xt, Nn, NHID, Nn, Nn, NHID, NHID, nullptr, dis, nullptr, 0, 1);
  // xw2 = X_tilde @ W_g2
  gemm_f32_wmma<false><<<dim3(1, Nn / 128), blk, 0, stream>>>(
      Xt, W_g2, xw2, Nn, NCLASS, NHID, NHID, NCLASS, NCLASS, nullptr, nullptr,
      nullptr, 0, 0);
  // z = adj @ xw2 + b_g2
  gemm_f32_wmma<false><<<dim3(1, Nn / 128), blk, 0, stream>>>(
      adj, xw2, zbuf, Nn, NCLASS, Nn, Nn, NCLASS, NCLASS, b_g2, nullptr,
      nullptr, 0, 0);
  // output softmax
  softmax16_kernel<<<Nn / 256, blk, 0, stream>>>(zbuf, out_probs);
}